// DMoN_32624571581048
// MI455X (gfx1250) — compile-verified
//
#include <hip/hip_runtime.h>
#include <hip/hip_bf16.h>
#include <math.h>

#define NODES 10000
#define IN_C 128
#define HID 256
#define KCL 16

typedef __attribute__((ext_vector_type(2))) float v2f;
typedef __attribute__((ext_vector_type(8))) float v8f;

// ---------------------------------------------------------------------------
// WMMA f32 GEMM: Out[M,Nd] = (RELU ? relu(A) : A)[M,Kd] @ W[Kd,Nd] (+ bias)
// One wave computes a 16 x (16*NT) strip: the A fragment (V_WMMA 16x4 f32
// layout) is loaded once per k-step and reused across NT accumulators,
// quadrupling WMMA density and cutting strided A traffic by NT.
// ---------------------------------------------------------------------------
template <int RELU, int NT>
__global__ void gemm_wmma_f32(const float* __restrict__ A,
                              const float* __restrict__ W,
                              const float* __restrict__ bias,
                              float* __restrict__ Out,
                              int M, int Kd, int Nd) {
    const int lane    = threadIdx.x & 31;
    const int waveId  = (blockIdx.x * (blockDim.x >> 5)) + (threadIdx.x >> 5);
    const int tilesN  = Nd >> 4;
    const int tilesNG = tilesN / NT;              // groups of NT adjacent N tiles
    const int nWaves  = (M >> 4) * tilesNG;
    if (waveId >= nWaves) return;                 // wave-uniform, EXEC stays all-1

    const int tm  = waveId / tilesNG;
    const int tg  = waveId % tilesNG;

    const int rlo = lane & 15;                    // row (A) / col (B,C,D) in tile
    const int hi  = lane >> 4;                    // 0: K+0/1, M+0..7 ; 1: K+2/3, M+8..15

    const int aRow  = tm * 16 + rlo;
    const int col0  = (tg * NT) * 16 + rlo;       // first column this lane touches

    const float* __restrict__ aPtr = A + (size_t)aRow * Kd;

    v8f acc[NT];
    #pragma unroll
    for (int t = 0; t < NT; ++t) acc[t] = (v8f){};

    for (int k0 = 0; k0 < Kd; k0 += 4) {
        const int kk = k0 + hi * 2;
        __builtin_prefetch(aPtr + k0 + 32, 0, 1);   // global_prefetch_b8 (speculative)
        float a0 = aPtr[kk];
        float a1 = aPtr[kk + 1];
        if (RELU) { a0 = fmaxf(a0, 0.0f); a1 = fmaxf(a1, 0.0f); }
        v2f a; a[0] = a0; a[1] = a1;

        const float* __restrict__ w0 = W + (size_t)kk * Nd + col0;
        const float* __restrict__ w1 = W + (size_t)(kk + 1) * Nd + col0;
        #pragma unroll
        for (int t = 0; t < NT; ++t) {
            v2f b;
            b[0] = w0[t * 16];
            b[1] = w1[t * 16];
            // (neg_a, A, neg_b, B, c_mod, C, reuse_a, reuse_b)
            acc[t] = __builtin_amdgcn_wmma_f32_16x16x4_f32(false, a, false, b,
                                                           (short)0, acc[t], false, false);
        }
    }

    #pragma unroll
    for (int t = 0; t < NT; ++t) {
        const int c = col0 + t * 16;
        float bv = 0.0f;
        if (bias) bv = bias[c];
        #pragma unroll
        for (int v = 0; v < 8; ++v) {
            const int r = tm * 16 + v + hi * 8;
            Out[(size_t)r * Nd + c] = acc[t][v] + bv;
        }
    }
}

// ---------------------------------------------------------------------------
// Degrees: GCN in-degree (dst side, + self loop added later) and pooling
// out-degree (src side, = adj row sums); also total edge weight (=> 2m).
// ---------------------------------------------------------------------------
__global__ void edge_degrees(const int* __restrict__ src, const int* __restrict__ dst,
                             const float* __restrict__ ew,
                             float* __restrict__ deg_gcn, float* __restrict__ deg_pool,
                             float* __restrict__ sum_ew, int E) {
    __shared__ float ls;
    if (threadIdx.x == 0) ls = 0.0f;
    __syncthreads();
    const int e = blockIdx.x * blockDim.x + threadIdx.x;
    float w = 0.0f;
    if (e < E) {
        w = ew[e];
        atomicAdd(&deg_gcn[dst[e]], w);
        atomicAdd(&deg_pool[src[e]], w);
    }
    atomicAdd(&ls, w);
    __syncthreads();
    if (threadIdx.x == 0) atomicAdd(sum_ew, ls);
}

// dis[i] = rsqrt(1 + weighted_in_degree)   (self-loop weight 1 => deg > 0 always)
__global__ void node_dis(const float* __restrict__ deg_gcn, float* __restrict__ dis, int n) {
    const int i = blockIdx.x * blockDim.x + threadIdx.x;
    if (i < n) dis[i] = rsqrtf(1.0f + deg_gcn[i]);
}

// Self-loop term + bias: Out[i,c] = dis[i]^2 * H[i,c] + b[c]
__global__ void agg_init(const float* __restrict__ H, const float* __restrict__ dis,
                         const float* __restrict__ bias, float* __restrict__ Out,
                         int n, int hid) {
    const int idx = blockIdx.x * blockDim.x + threadIdx.x;
    if (idx >= n * hid) return;
    const int i = idx / hid;
    const int c = idx - i * hid;
    const float d = dis[i];
    Out[idx] = d * d * H[idx] + bias[c];
}

// Edge scatter: Out[dst] += dis[src]*ew*dis[dst] * H[src]; float4 gather,
// 4 global fp atomics per thread. 64 threads per edge (hid=256).
__global__ void agg_edges(const float* __restrict__ H, const float* __restrict__ dis,
                          const int* __restrict__ src, const int* __restrict__ dst,
                          const float* __restrict__ ew, float* __restrict__ Out,
                          int E, int hid) {
    const int gpe = hid >> 2;
    const int tid = blockIdx.x * blockDim.x + threadIdx.x;
    if (tid >= E * gpe) return;
    const int e = tid / gpe;
    const int g = tid - e * gpe;
    const int sN = src[e], dN = dst[e];
    const float nrm = dis[sN] * ew[e] * dis[dN];
    const float4 h4 = *(const float4*)&H[(size_t)sN * hid + g * 4];
    float* o = &Out[(size_t)dN * hid + g * 4];
    atomicAdd(o + 0, nrm * h4.x);
    atomicAdd(o + 1, nrm * h4.y);
    atomicAdd(o + 2, nrm * h4.z);
    atomicAdd(o + 3, nrm * h4.w);
}

// Row softmax over K=16 clusters (in place on the s region of d_out)
__global__ void softmax16(float* __restrict__ s, int n) {
    const int i = blockIdx.x * blockDim.x + threadIdx.x;
    if (i >= n) return;
    float v[KCL];
    float mx = -1e30f;
    #pragma unroll
    for (int k = 0; k < KCL; ++k) { v[k] = s[(size_t)i * KCL + k]; mx = fmaxf(mx, v[k]); }
    float sum = 0.0f;
    #pragma unroll
    for (int k = 0; k < KCL; ++k) { v[k] = __expf(v[k] - mx); sum += v[k]; }
    const float inv = 1.0f / sum;
    #pragma unroll
    for (int k = 0; k < KCL; ++k) s[(size_t)i * KCL + k] = v[k] * inv;
}

// red[0..15]  += deg_pool[i]*s[i,k]   (c vector)
// red[16..31] += s[i,k]               (cluster sizes)
__global__ void reduce_nodes(const float* __restrict__ s, const float* __restrict__ deg_pool,
                             float* __restrict__ red, int n) {
    __shared__ float sc[32];
    if (threadIdx.x < 32) sc[threadIdx.x] = 0.0f;
    __syncthreads();
    const int i = blockIdx.x * blockDim.x + threadIdx.x;
    if (i < n) {
        const float dp = deg_pool[i];
        #pragma unroll
        for (int k = 0; k < KCL; ++k) {
            const float sv = s[(size_t)i * KCL + k];
            atomicAdd(&sc[k], dp * sv);
            atomicAdd(&sc[16 + k], sv);
        }
    }
    __syncthreads();
    if (threadIdx.x < 32) atomicAdd(&red[threadIdx.x], sc[threadIdx.x]);
}

// red[32] += ew_e * dot(s[src_e], s[dst_e])   => trace(S^T A S)
__global__ void edge_trace(const float* __restrict__ s, const int* __restrict__ src,
                           const int* __restrict__ dst, const float* __restrict__ ew,
                           float* __restrict__ red, int E) {
    __shared__ float ls;
    if (threadIdx.x == 0) ls = 0.0f;
    __syncthreads();
    const int e = blockIdx.x * blockDim.x + threadIdx.x;
    float t = 0.0f;
    if (e < E) {
        const int a = src[e], b = dst[e];
        float dot = 0.0f;
        #pragma unroll
        for (int k = 0; k < KCL; ++k)
            dot += s[(size_t)a * KCL + k] * s[(size_t)b * KCL + k];
        t = ew[e] * dot;
    }
    atomicAdd(&ls, t);
    __syncthreads();
    if (threadIdx.x == 0) atomicAdd(&red[32], ls);
}

// Scalars: loss, 100*spectral, 100*cluster
__global__ void finalize(const float* __restrict__ red, float* __restrict__ out3, float n) {
    if (threadIdx.x != 0 || blockIdx.x != 0) return;
    const float two_m = red[33];                 // sum of all edge weights = 2m
    float c2 = 0.0f, cs2 = 0.0f;
    #pragma unroll
    for (int k = 0; k < KCL; ++k) {
        c2  += red[k] * red[k];
        cs2 += red[16 + k] * red[16 + k];
    }
    const float t1 = red[32];
    const float spectral = -(t1 - c2 / two_m) / two_m;
    const float cluster  = sqrtf(cs2) * 4.0f / n - 1.0f;   // sqrt(K)=4
    const float spec100 = 100.0f * spectral;
    const float clus100 = 100.0f * cluster;
    out3[0] = spec100 + clus100;
    out3[1] = spec100;
    out3[2] = clus100;
}

extern "C" void kernel_launch(void* const* d_in, const int* in_sizes, int n_in,
                              void* d_out, int out_size, void* d_ws, size_t ws_size,
                              hipStream_t stream) {
    const float* x   = (const float*)d_in[0];
    const int*  eix  = (const int*)d_in[1];
    const float* ew  = (const float*)d_in[2];
    const float* W1  = (const float*)d_in[3];
    const float* b1  = (const float*)d_in[4];
    const float* W2  = (const float*)d_in[5];
    const float* b2  = (const float*)d_in[6];
    const float* Wp  = (const float*)d_in[7];
    const float* bp  = (const float*)d_in[8];

    const int E = in_sizes[2];          // 160000
    const int n = NODES;                // 10000

    const int* src = eix;               // edge_index[0]
    const int* dst = eix + E;           // edge_index[1]

    // Workspace layout
    float* bufA     = (float*)d_ws;                     // N*HID (GEMM out)
    float* bufB     = bufA + (size_t)n * HID;           // N*HID (aggregated)
    float* deg_gcn  = bufB + (size_t)n * HID;           // N
    float* deg_pool = deg_gcn + n;                      // N
    float* dis      = deg_pool + n;                     // N
    float* red      = dis + n;                          // 64 scalars

    float* s_out  = (float*)d_out;                      // [N,K]
    float* out3   = s_out + (size_t)n * KCL;            // loss, spec, clus

    // Zero accumulators (deg_gcn, deg_pool, dis(overwritten anyway), red)
    hipMemsetAsync(deg_gcn, 0, (size_t)(3 * n + 64) * sizeof(float), stream);

    const int TB = 256;

    // Degrees + 2m
    edge_degrees<<<(E + TB - 1) / TB, TB, 0, stream>>>(src, dst, ew, deg_gcn, deg_pool, &red[33], E);
    node_dis<<<(n + TB - 1) / TB, TB, 0, stream>>>(deg_gcn, dis, n);

    const int wavesPerBlk = TB / 32;
    const int wavesL      = (n / 16) * (HID / 16) / 4;   // NT=4 -> 2500 waves
    const int wavesP      = (n / 16) * (KCL / 16);       // NT=1 -> 625 waves
    const int blksL       = (wavesL + wavesPerBlk - 1) / wavesPerBlk;
    const int blksP       = (wavesP + wavesPerBlk - 1) / wavesPerBlk;

    // Layer 1: H = x @ W1 ; aggregate ; (relu deferred to next GEMM load)
    gemm_wmma_f32<0, 4><<<blksL, TB, 0, stream>>>(x, W1, nullptr, bufA, n, IN_C, HID);
    agg_init<<<(n * HID + TB - 1) / TB, TB, 0, stream>>>(bufA, dis, b1, bufB, n, HID);
    agg_edges<<<(E * (HID / 4) + TB - 1) / TB, TB, 0, stream>>>(bufA, dis, src, dst, ew, bufB, E, HID);

    // Layer 2: H = relu(h1) @ W2 ; aggregate
    gemm_wmma_f32<1, 4><<<blksL, TB, 0, stream>>>(bufB, W2, nullptr, bufA, n, HID, HID);
    agg_init<<<(n * HID + TB - 1) / TB, TB, 0, stream>>>(bufA, dis, b2, bufB, n, HID);
    agg_edges<<<(E * (HID / 4) + TB - 1) / TB, TB, 0, stream>>>(bufA, dis, src, dst, ew, bufB, E, HID);

    // Pooling assignments: logits = relu(h2) @ Wp + bp -> softmax -> s (in d_out)
    gemm_wmma_f32<1, 1><<<blksP, TB, 0, stream>>>(bufB, Wp, bp, s_out, n, HID, KCL);
    softmax16<<<(n + TB - 1) / TB, TB, 0, stream>>>(s_out, n);

    // Scalar reductions (never materialize dense adj)
    reduce_nodes<<<(n + TB - 1) / TB, TB, 0, stream>>>(s_out, deg_pool, red, n);
    edge_trace<<<(E + TB - 1) / TB, TB, 0, stream>>>(s_out, src, dst, ew, red, E);
    finalize<<<1, 64, 0, stream>>>(red, out3, (float)n);
}